// sqgkt_5858335392061
// MI455X (gfx1250) — compile-verified
//
#include <hip/hip_runtime.h>
#include <hip/hip_bf16.h>

typedef __attribute__((ext_vector_type(2))) float v2f;
typedef __attribute__((ext_vector_type(8))) float v8f;

constexpr int DD  = 100;   // embedding dim
constexpr int STc = 101;   // LDS row stride (odd -> conflict-free)
constexpr int PP  = 199;   // T-1
constexpr int BB  = 16;
constexpr int TT  = 200;
constexpr int NBP = BB * PP;   // 3184

__device__ inline v8f wmma4(v2f a, v2f b, v8f c) {
  // D = A(16x4,f32) * B(4x16,f32) + C(16x16,f32)
  return __builtin_amdgcn_wmma_f32_16x16x4_f32(false, a, false, b, (short)0, c, false, false);
}

// ---------------------------------------------------------------------------
// Aggregation matmul helper: rows x 100 (LDS, padded) times W (100x100, global)
// Out-of-range rows/cols are computed on clamped addresses (no branches) and
// simply not written back, so the WMMA inner loop is branch-free.
// ---------------------------------------------------------------------------
__device__ inline void agg_mm(const float* sb, const float* __restrict__ W,
                              const float* __restrict__ bias, int rows,
                              float* dstL, float* dstG) {
  const int tid  = threadIdx.x;
  const int wave = tid >> 5, nw = blockDim.x >> 5;
  const int lane = tid & 31, l15 = lane & 15;
  const int koff = (lane >= 16) ? 2 : 0;
  const int mt_n = (rows + 15) >> 4;
  const int total = mt_n * 7;                    // 7 N-tiles cover 100 cols
  for (int tile = wave; tile < total; tile += nw) {
    const int mt = tile / 7, nt = tile % 7;
    const int col = nt * 16 + l15;
    const int colc = (col < DD) ? col : (DD - 1);      // clamp, no branchy loads
    v8f acc = {};
    const float* sA = sb + (mt * 16 + l15) * STc + koff;
    const float* Wp = W + (long long)koff * DD + colc;
#pragma unroll 5
    for (int kk = 0; kk < 25; ++kk) {
      v2f a, b;
      a.x = sA[kk * 4];
      a.y = sA[kk * 4 + 1];
      b.x = Wp[(long long)(kk * 4) * DD];
      b.y = Wp[(long long)(kk * 4 + 1) * DD];
      acc = wmma4(a, b, acc);
    }
    if (col < DD) {
      const float bv = bias[col];
      const int rbase = mt * 16 + ((lane >= 16) ? 8 : 0);
      for (int r = 0; r < 8; ++r) {
        const int rr = rbase + r;
        if (rr < rows) {
          const float v = tanhf(acc[r] + bv);
          if (dstG) dstG[rr * DD + col] = v;
          else      dstL[rr * STc + col] = v;
        }
      }
    }
  }
}

// ---------------------------------------------------------------------------
// Aggregation: one block per (b,t,side). Tree (1/4/20/80 rows x 100) in LDS,
// triangular cascade of DxD matmuls via V_WMMA_F32_16X16X4_F32.
// ---------------------------------------------------------------------------
__global__ __launch_bounds__(256) void agg_kernel(
    const int* __restrict__ question, const int* __restrict__ user,
    const int* __restrict__ q_neighbors, const int* __restrict__ s_neighbors,
    const int* __restrict__ u_neighbors, const int* __restrict__ q_neighbors_2,
    const float* __restrict__ emb_q, const float* __restrict__ emb_s,
    const float* __restrict__ emb_u, const float* __restrict__ emb_q2,
    const float* __restrict__ agg_W, const float* __restrict__ agg_b,
    const float* __restrict__ W_last, const float* __restrict__ b_last,
    float* __restrict__ agg_out_q, float* __restrict__ agg_out_u) {
  __shared__ float ebuf[105 * STc];   // rows: e0@0, e1@1..4, e2@5..24, e3@25..104
  __shared__ float sbuf[32 * STc];
  __shared__ int ridx[105];
  const int bp = blockIdx.x, side = blockIdx.y;
  const int b = bp / PP, t = bp % PP;
  const int tid = threadIdx.x;

  const int *tab1, *tab2, *tab3;
  const float *embA, *embB;
  int root;
  if (side == 0) {
    root = question[b * TT + t];
    tab1 = q_neighbors; tab2 = s_neighbors; tab3 = q_neighbors;
    embA = emb_q; embB = emb_s;
  } else {
    root = user[b * TT + t];
    tab1 = u_neighbors; tab2 = q_neighbors_2; tab3 = u_neighbors;
    embA = emb_u; embB = emb_q2;
  }
  if (tid == 0) ridx[0] = root;
  if (tid < 4)  ridx[1 + tid] = tab1[root * 4 + tid];
  __syncthreads();
  if (tid < 20) ridx[5 + tid] = tab2[ridx[1 + tid / 5] * 5 + (tid % 5)];
  __syncthreads();
  if (tid < 80) ridx[25 + tid] = tab3[ridx[5 + tid / 4] * 4 + (tid % 4)];
  __syncthreads();
  for (int e = tid; e < 105 * DD; e += blockDim.x) {
    const int row = e / DD, col = e % DD;
    const float* src = (row == 0 || (row >= 5 && row < 25)) ? embA : embB;
    ebuf[row * STc + col] = src[(long long)ridx[row] * DD + col];
  }
  __syncthreads();

  const int offs[4]  = {0, 1, 5, 25};
  const int rowsA[3] = {1, 4, 20};
  const int child[3] = {4, 5, 4};
  for (int i = 0; i < 3; ++i) {
    for (int j = 0; j < 3 - i; ++j) {
      const int rows = rowsA[j], cc = child[j], off = offs[j], offc = offs[j + 1];
      const float inv = 1.f / (float)cc;
      const int mp = ((rows + 15) >> 4) << 4;
      for (int e = tid; e < mp * DD; e += blockDim.x) {
        const int r = e / DD, col = e % DD;
        float v = 0.f;
        if (r < rows) {
          float s = 0.f;
          for (int k = 0; k < cc; ++k) s += ebuf[(offc + r * cc + k) * STc + col];
          v = ebuf[(off + r) * STc + col] + s * inv;
        }
        sbuf[r * STc + col] = v;
      }
      __syncthreads();
      agg_mm(sbuf, agg_W + j * DD * DD, agg_b + j * DD, rows, &ebuf[off * STc], nullptr);
      __syncthreads();
    }
  }
  // final: tanh(e0 @ W_last + b_last) -> global
  for (int e = tid; e < 16 * DD; e += blockDim.x) {
    const int r = e / DD, col = e % DD;
    sbuf[r * STc + col] = (r == 0) ? ebuf[col] : 0.f;
  }
  __syncthreads();
  float* outp = (side == 0 ? agg_out_q : agg_out_u) + (long long)bp * DD;
  agg_mm(sbuf, W_last, b_last, 1, nullptr, outp);
}

// ---------------------------------------------------------------------------
// Generic batched WMMA-f32 GEMM: C = A(MxK) * B(KxN or NxK if transB) + bias.
// One wave per 16x16 output tile; K chunked by 4 (K must be a multiple of 4).
// Edge tiles use clamped addresses (branch-free loads); writeback is guarded.
// ---------------------------------------------------------------------------
__global__ __launch_bounds__(32) void gemm_k(
    const float* __restrict__ A, int lda, long long sA,
    const float* __restrict__ Bm, int ldb, long long sB, int transB,
    const float* __restrict__ bias,
    float* __restrict__ C, int ldc, long long sC,
    int M, int N, int K) {
  const int batch = blockIdx.z;
  A  += (long long)batch * sA;
  Bm += (long long)batch * sB;
  C  += (long long)batch * sC;
  const int lane = threadIdx.x & 31, l15 = lane & 15;
  const int koff = (lane >= 16) ? 2 : 0;
  const int row = blockIdx.y * 16 + l15;
  const int col = blockIdx.x * 16 + l15;
  const int rowc = (row < M) ? row : (M - 1);
  const int colc = (col < N) ? col : (N - 1);
  v8f acc = {};
  const int kt = K >> 2;
  const float* Ap = A + (long long)rowc * lda + koff;
  const float* Bp;
  long long bstride;   // stride between consecutive k for B access
  if (transB) { Bp = Bm + (long long)colc * ldb + koff; bstride = 1; }
  else        { Bp = Bm + (long long)koff * ldb + colc; bstride = ldb; }
#pragma unroll 5
  for (int kk = 0; kk < kt; ++kk) {
    const int k0 = kk * 4;
    v2f a, b;
    a.x = Ap[k0];
    a.y = Ap[k0 + 1];
    b.x = Bp[(long long)k0 * bstride];
    b.y = Bp[(long long)(k0 + 1) * bstride];
    acc = wmma4(a, b, acc);
  }
  if (col >= N) return;
  const float bv = bias ? bias[col] : 0.f;
  const int rbase = blockIdx.y * 16 + ((lane >= 16) ? 8 : 0);
  for (int r = 0; r < 8; ++r) {
    const int rr = rbase + r;
    if (rr < M) C[(long long)rr * ldc + col] = acc[r] + bv;
  }
}

// ---------------------------------------------------------------------------
// Elementwise / small kernels
// ---------------------------------------------------------------------------
__global__ void build_x_kernel(const int* __restrict__ question, const int* __restrict__ mask,
                               const float* __restrict__ agg_q, const float* __restrict__ agg_u,
                               const float* __restrict__ emb_q, const float* __restrict__ emb_q2,
                               float* __restrict__ X) {
  const int idx = blockIdx.x * blockDim.x + threadIdx.x;
  if (idx >= NBP * 2 * DD) return;
  const int c = idx % (2 * DD);
  const int bp = idx / (2 * DD);
  const int b = bp / PP, t = bp % PP;
  const int m = mask[b * TT + t];
  const int q = question[b * TT + t];
  float v;
  if (c < DD) v = (m == 1) ? agg_q[(long long)bp * DD + c] : emb_q[(long long)q * DD + c];
  else {
    const int c2 = c - DD;
    v = (m == 1) ? agg_u[(long long)bp * DD + c2] : emb_q2[(long long)q * DD + c2];
  }
  X[(long long)bp * 2 * DD + c] = v;
}

__global__ __launch_bounds__(128) void softmax_li_kernel(
    const float* __restrict__ X, const float* __restrict__ XL,
    const int* __restrict__ response, const float* __restrict__ emb_resp,
    float* __restrict__ LI) {
  __shared__ float red[128];
  const int bp = blockIdx.x, tid = threadIdx.x;
  const float* xl = XL + (long long)bp * 200;
  const float* x  = X  + (long long)bp * 200;
  float mx = -1e30f;
  for (int c = tid; c < 200; c += 128) mx = fmaxf(mx, xl[c]);
  red[tid] = mx; __syncthreads();
  for (int s = 64; s > 0; s >>= 1) { if (tid < s) red[tid] = fmaxf(red[tid], red[tid + s]); __syncthreads(); }
  mx = red[0]; __syncthreads();
  float sm = 0.f;
  for (int c = tid; c < 200; c += 128) sm += __expf(xl[c] - mx);
  red[tid] = sm; __syncthreads();
  for (int s = 64; s > 0; s >>= 1) { if (tid < s) red[tid] += red[tid + s]; __syncthreads(); }
  const float inv = 1.f / red[0];
  float* li = LI + (long long)bp * 300;
  for (int c = tid; c < 200; c += 128) li[c] = x[c] * __expf(xl[c] - mx) * inv;
  const int b = bp / PP, t = bp % PP;
  const int r = response[b * TT + t];
  for (int c = tid; c < DD; c += 128) li[200 + c] = emb_resp[r * DD + c];
}

__global__ __launch_bounds__(128) void h_kernel(const float* __restrict__ gates,
                                                const float* __restrict__ w_pred,
                                                float* __restrict__ h, float* __restrict__ ah) {
  __shared__ float red[128];
  const int bp = blockIdx.x, tid = threadIdx.x;
  const float* g = gates + (long long)bp * 400;
  float part = 0.f;
  if (tid < DD) {
    const float ig = g[tid], gg = g[200 + tid], og = g[300 + tid];
    const float si = 1.f / (1.f + __expf(-ig));
    const float so = 1.f / (1.f + __expf(-og));
    const float hv = so * tanhf(si * tanhf(gg));
    h[(long long)bp * DD + tid] = hv;
    part = hv * w_pred[DD + tid];
  }
  red[tid] = part; __syncthreads();
  for (int s = 64; s > 0; s >>= 1) { if (tid < s) red[tid] += red[tid + s]; __syncthreads(); }
  if (tid == 0) ah[bp] = red[0];
}

__global__ void qsc_kernel(const int* __restrict__ question, const int* __restrict__ qs_idx,
                           const float* __restrict__ emb_q, const float* __restrict__ emb_s,
                           float* __restrict__ qsc) {
  const long long idx = (long long)blockIdx.x * blockDim.x + threadIdx.x;
  if (idx >= (long long)NBP * 5 * DD) return;
  const int col = (int)(idx % DD);
  const long long rr = idx / DD;
  const int m = (int)(rr % 5);
  const int bp = (int)(rr / 5);
  const int b = bp / PP, t = bp % PP;
  const int qn = question[b * TT + t + 1];
  const float* src;
  long long row;
  if (m == 0) { src = emb_q; row = qn; }
  else        { src = emb_s; row = qs_idx[qn * 4 + (m - 1)]; }
  qsc[idx] = src[row * DD + col];
}

__global__ void aq_l0_kernel(const float* __restrict__ qsc, const float* __restrict__ Qq,
                             const float* __restrict__ w_pred, const float* __restrict__ Wk_b,
                             float* __restrict__ aq, float* __restrict__ l0) {
  const int idx = blockIdx.x * blockDim.x + threadIdx.x;
  if (idx >= NBP * 5) return;
  const float* qr = qsc + (long long)idx * DD;
  const float* Qr = Qq  + (long long)idx * DD;
  float a = 0.f, l = 0.f;
  for (int d = 0; d < DD; ++d) { a += qr[d] * w_pred[d]; l += Qr[d] * Wk_b[d]; }
  aq[idx] = a; l0[idx] = l;
}

__global__ __launch_bounds__(256) void pred_kernel(
    const int* __restrict__ question, const int* __restrict__ qs_idx,
    const float* __restrict__ logits, const float* __restrict__ l0,
    const float* __restrict__ aq, const float* __restrict__ ah,
    const float* __restrict__ b_pred, float* __restrict__ out) {
  __shared__ float lsh[5 * PP];
  __shared__ float red[256];
  __shared__ int validsh[PP];
  const int bp = blockIdx.x, tid = threadIdx.x;
  const int b = bp / PP, t = bp % PP;
  const int qn = question[b * TT + t + 1];
  const int sk0 = qs_idx[qn * 4 + 0], sk1 = qs_idx[qn * 4 + 1];
  const int sk2 = qs_idx[qn * 4 + 2], sk3 = qs_idx[qn * 4 + 3];
  for (int n = tid; n < PP; n += 256) {
    const int qh = question[b * TT + n];
    int v = 0;
    for (int j = 0; j < 4; ++j) {
      const int s = qs_idx[qh * 4 + j];
      v |= (s == sk0) | (s == sk1) | (s == sk2) | (s == sk3);
    }
    validsh[n] = (n == t) || (n < t && v);
  }
  __syncthreads();
  const float bpv = b_pred[0];
  const float* lrow = logits + (long long)b * (PP * 5 * PP) + (long long)(t * 5) * PP;
  float mx = -1e30f;
  for (int i = tid; i < 5 * PP; i += 256) {
    const int m = i / PP, n = i % PP;
    float lg = lrow[(long long)m * PP + n];
    if (n == 0 && t >= 1) lg = l0[bp * 5 + m];
    if (!validsh[n]) lg = -1e9f;
    lsh[i] = lg;
    mx = fmaxf(mx, lg);
  }
  red[tid] = mx; __syncthreads();
  for (int s = 128; s > 0; s >>= 1) { if (tid < s) red[tid] = fmaxf(red[tid], red[tid + s]); __syncthreads(); }
  mx = red[0]; __syncthreads();
  float se = 0.f, sv = 0.f;
  for (int i = tid; i < 5 * PP; i += 256) {
    const int m = i / PP, n = i % PP;
    const float e = __expf(lsh[i] - mx);
    const float ahv = (n == 0 && t >= 1) ? 0.f : ah[b * PP + n];
    const float val = 1.f / (1.f + __expf(-(aq[bp * 5 + m] + ahv + bpv)));
    se += e; sv += e * val;
  }
  red[tid] = se; __syncthreads();
  for (int s = 128; s > 0; s >>= 1) { if (tid < s) red[tid] += red[tid + s]; __syncthreads(); }
  se = red[0]; __syncthreads();
  red[tid] = sv; __syncthreads();
  for (int s = 128; s > 0; s >>= 1) { if (tid < s) red[tid] += red[tid + s]; __syncthreads(); }
  if (tid == 0) {
    out[b * TT + t + 1] = red[0] / se;
    if (t == 0) out[b * TT] = 0.5f;
  }
}

// ---------------------------------------------------------------------------
extern "C" void kernel_launch(void* const* d_in, const int* in_sizes, int n_in,
                              void* d_out, int out_size, void* d_ws, size_t ws_size,
                              hipStream_t stream) {
  const int* user          = (const int*)d_in[0];
  const int* question      = (const int*)d_in[1];
  const int* response      = (const int*)d_in[2];
  const int* mask          = (const int*)d_in[3];
  const int* q_neighbors   = (const int*)d_in[4];
  const int* s_neighbors   = (const int*)d_in[5];
  const int* u_neighbors   = (const int*)d_in[6];
  const int* q_neighbors_2 = (const int*)d_in[7];
  const int* qs_idx        = (const int*)d_in[8];
  const float* emb_q    = (const float*)d_in[9];
  const float* emb_s    = (const float*)d_in[10];
  const float* emb_u    = (const float*)d_in[11];
  const float* emb_q2   = (const float*)d_in[12];
  const float* emb_resp = (const float*)d_in[13];
  const float* W_ih   = (const float*)d_in[14];
  const float* b_ih   = (const float*)d_in[15];
  const float* agg_W  = (const float*)d_in[16];
  const float* agg_b  = (const float*)d_in[17];
  const float* W_last = (const float*)d_in[18];
  const float* b_last = (const float*)d_in[19];
  const float* att1_W = (const float*)d_in[20];
  const float* att1_b = (const float*)d_in[21];
  const float* Wq_w   = (const float*)d_in[22];
  const float* Wq_b   = (const float*)d_in[23];
  const float* Wk_w   = (const float*)d_in[24];
  const float* Wk_b   = (const float*)d_in[25];
  const float* w_pred = (const float*)d_in[26];
  const float* b_pred = (const float*)d_in[27];
  float* out = (float*)d_out;

  float* ws = (float*)d_ws;
  long long o = 0;
  float* aggq  = ws + o; o += (long long)NBP * 100;
  float* aggu  = ws + o; o += (long long)NBP * 100;
  float* X     = ws + o; o += (long long)NBP * 200;
  float* XL    = ws + o; o += (long long)NBP * 200;
  float* LI    = ws + o; o += (long long)NBP * 300;
  float* gates = ws + o; o += (long long)NBP * 400;
  float* h     = ws + o; o += (long long)NBP * 100;
  float* Kh    = ws + o; o += (long long)NBP * 100;
  float* ah    = ws + o; o += (long long)NBP;
  float* qsc   = ws + o; o += (long long)NBP * 500;
  float* Qq    = ws + o; o += (long long)NBP * 500;
  float* aq    = ws + o; o += (long long)NBP * 5;
  float* l0    = ws + o; o += (long long)NBP * 5;
  float* logits = ws + o; o += 16LL * 995 * 199;
  (void)in_sizes; (void)n_in; (void)out_size; (void)ws_size;

  // 1) graph aggregation (q side + u side)
  agg_kernel<<<dim3(NBP, 2), 256, 0, stream>>>(
      question, user, q_neighbors, s_neighbors, u_neighbors, q_neighbors_2,
      emb_q, emb_s, emb_u, emb_q2, agg_W, agg_b, W_last, b_last, aggq, aggu);

  // 2) x = concat(eq1, eq2)
  build_x_kernel<<<(NBP * 200 + 255) / 256, 256, 0, stream>>>(
      question, mask, aggq, aggu, emb_q, emb_q2, X);

  // 3) XL = X @ att1_W + att1_b   (M=3184, K=200, N=200)
  gemm_k<<<dim3(13, 199, 1), 32, 0, stream>>>(X, 200, 0, att1_W, 200, 0, 0,
                                              att1_b, XL, 200, 0, NBP, 200, 200);

  // 4) x *= softmax(XL); LI = [x, emb_resp[r]]
  softmax_li_kernel<<<NBP, 128, 0, stream>>>(X, XL, response, emb_resp, LI);

  // 5) gates = LI @ W_ih^T + b_ih  (M=3184, K=300, N=400, B transposed)
  gemm_k<<<dim3(25, 199, 1), 32, 0, stream>>>(LI, 300, 0, W_ih, 300, 0, 1,
                                              b_ih, gates, 400, 0, NBP, 400, 300);

  // 6) h + ah
  h_kernel<<<NBP, 128, 0, stream>>>(gates, w_pred, h, ah);

  // 7) Kh = h @ Wk_w + Wk_b  (M=3184, K=100, N=100)
  gemm_k<<<dim3(7, 199, 1), 32, 0, stream>>>(h, 100, 0, Wk_w, 100, 0, 0,
                                             Wk_b, Kh, 100, 0, NBP, 100, 100);

  // 8) qs_concat gather
  qsc_kernel<<<(int)(((long long)NBP * 500 + 255) / 256), 256, 0, stream>>>(
      question, qs_idx, emb_q, emb_s, qsc);

  // 9) Qq = qsc @ Wq_w + Wq_b  (M=15920, K=100, N=100)
  gemm_k<<<dim3(7, 995, 1), 32, 0, stream>>>(qsc, 100, 0, Wq_w, 100, 0, 0,
                                             Wq_b, Qq, 100, 0, NBP * 5, 100, 100);

  // 10) aq, logits0
  aq_l0_kernel<<<(NBP * 5 + 255) / 256, 256, 0, stream>>>(qsc, Qq, w_pred, Wk_b, aq, l0);

  // 11) logits[b] = Qq_b (995x100) @ Kh_b^T (100x199), batched over b
  gemm_k<<<dim3(13, 63, 16), 32, 0, stream>>>(
      Qq, 100, 995LL * 100, Kh, 100, (long long)PP * 100, 1,
      nullptr, logits, PP, 995LL * 199, 995, PP, 100);

  // 12) masked softmax + sigmoid-value reduction -> preds
  pred_kernel<<<NBP, 256, 0, stream>>>(question, qs_idx, logits, l0, aq, ah, b_pred, out);
}